// GCNEncoder_15006615732583
// MI455X (gfx1250) — compile-verified
//
#include <hip/hip_runtime.h>
#include <math.h>

typedef __attribute__((ext_vector_type(2))) float v2f;
typedef __attribute__((ext_vector_type(8))) float v8f;

#define NN 50000
#define NE 800000

// ---------------------------------------------------------------------------
// GEMM: H[M,N] = X[M,K] @ W[K,N], fp32 via V_WMMA_F32_16X16X4_F32.
// grid.x = M/16 tiles; blockDim.x = (N/16)*32 (one wave per 16-col tile).
// A frag (16x4 f32): lanes 0-15 -> K={0,1}, lanes 16-31 -> K={2,3} (ISA 7.12.2)
// B frag (4x16 f32): row-striped across lanes, mirrored K split.
// C/D  (16x16 f32): VGPR r: lanes0-15 M=r, lanes16-31 M=r+8.
// ---------------------------------------------------------------------------
__global__ void gemm_wmma_f32(const float* __restrict__ X,
                              const float* __restrict__ W,
                              float* __restrict__ H,
                              int K, int N) {
  const int lane = threadIdx.x & 31;
  const int wave = threadIdx.x >> 5;
  const int m0   = blockIdx.x * 16;
  const int n0   = wave * 16;
  const int r    = lane & 15;
  const int khi  = (lane >> 4) << 1;  // 0 for lanes 0-15, 2 for lanes 16-31

  const float* aptr = X + (size_t)(m0 + r) * K + khi;
  const float* bptr = W + (size_t)khi * N + n0 + r;

  v8f acc = {0.f, 0.f, 0.f, 0.f, 0.f, 0.f, 0.f, 0.f};
  #pragma unroll 8
  for (int k = 0; k < K; k += 4) {
    v2f a, b;
    a.x = aptr[k];
    a.y = aptr[k + 1];
    b.x = bptr[(size_t)k * N];
    b.y = bptr[(size_t)(k + 1) * N];
    acc = __builtin_amdgcn_wmma_f32_16x16x4_f32(false, a, false, b,
                                                (short)0, acc, false, false);
  }

  const int crow = m0 + ((lane >> 4) << 3);
  const int ccol = n0 + r;
  #pragma unroll
  for (int i = 0; i < 8; ++i)
    H[(size_t)(crow + i) * N + ccol] = acc[i];
}

// ---------------------------------------------------------------------------
// Degree / normalization
// ---------------------------------------------------------------------------
__global__ void deg_init(float* __restrict__ deg) {
  int i = blockIdx.x * blockDim.x + threadIdx.x;
  if (i < NN) deg[i] = 1.0f;  // self-loop contributes 1 to every node
}

__global__ void deg_accum(const int* __restrict__ dst, float* __restrict__ deg) {
  int e = blockIdx.x * blockDim.x + threadIdx.x;
  if (e < NE) atomicAdd(&deg[dst[e]], 1.0f);
}

__global__ void deg_to_dinv(float* __restrict__ deg) {
  int i = blockIdx.x * blockDim.x + threadIdx.x;
  if (i < NN) {
    float d = deg[i];
    deg[i] = (d > 0.0f) ? rsqrtf(d) : 0.0f;
  }
}

// ---------------------------------------------------------------------------
// out[n,f] = h[n,f] * dinv[n]^2   (self-loop term; also zero-initializes out)
// ---------------------------------------------------------------------------
__global__ void self_loop_init(const float* __restrict__ h,
                               const float* __restrict__ dinv,
                               float* __restrict__ out, int F) {
  size_t idx = (size_t)blockIdx.x * blockDim.x + threadIdx.x;
  size_t total = (size_t)NN * F;
  if (idx < total) {
    int n = (int)(idx / F);
    float di = dinv[n];
    out[idx] = h[idx] * di * di;
  }
}

// ---------------------------------------------------------------------------
// Edge scatter: out[dst,f] += h[src,f] * dinv[src]*dinv[dst]
// One thread per (edge, feature); coalesced gather + coalesced f32 atomics.
// ---------------------------------------------------------------------------
__global__ void edge_scatter(const float* __restrict__ h,
                             const float* __restrict__ dinv,
                             const int* __restrict__ src,
                             const int* __restrict__ dst,
                             float* __restrict__ out,
                             int F, int logF) {
  size_t idx = (size_t)blockIdx.x * blockDim.x + threadIdx.x;
  size_t total = (size_t)NE * F;
  if (idx < total) {
    int e = (int)(idx >> logF);
    int f = (int)(idx & (size_t)(F - 1));
    int s = src[e];
    int d = dst[e];
    float norm = dinv[s] * dinv[d];
    atomicAdd(&out[(size_t)d * F + f], h[(size_t)s * F + f] * norm);
  }
}

// ---------------------------------------------------------------------------
// Epilogue: out += bias; optional sigmoid
// ---------------------------------------------------------------------------
__global__ void bias_act(float* __restrict__ out, const float* __restrict__ b,
                         int F, int do_sigmoid) {
  size_t idx = (size_t)blockIdx.x * blockDim.x + threadIdx.x;
  size_t total = (size_t)NN * F;
  if (idx < total) {
    float v = out[idx] + b[idx & (size_t)(F - 1)];
    if (do_sigmoid) v = 1.0f / (1.0f + __expf(-v));
    out[idx] = v;
  }
}

// ---------------------------------------------------------------------------
// Launch
// ---------------------------------------------------------------------------
extern "C" void kernel_launch(void* const* d_in, const int* in_sizes, int n_in,
                              void* d_out, int out_size, void* d_ws, size_t ws_size,
                              hipStream_t stream) {
  const float* x  = (const float*)d_in[0];          // [50000,128]
  const int*   eg = (const int*)d_in[1];            // [2,800000]
  const float* W0 = (const float*)d_in[2];          // [128,128]
  const float* b0 = (const float*)d_in[3];          // [128]
  const float* W1 = (const float*)d_in[4];          // [128,64]
  const float* b1 = (const float*)d_in[5];          // [64]
  const int* src = eg;
  const int* dst = eg + NE;

  float* dinv = (float*)d_ws;                       // 50000 floats
  float* bufA = dinv + 50048;                       // 50000*128 floats (h0, later h1)
  float* bufB = bufA + (size_t)NN * 128;            // 50000*128 floats (agg0)
  float* out  = (float*)d_out;                      // 50000*64 floats

  const int B = 256;

  // normalization coefficients (deg from dst incl. self-loops)
  deg_init   <<<(NN + B - 1) / B, B, 0, stream>>>(dinv);
  deg_accum  <<<(NE + B - 1) / B, B, 0, stream>>>(dst, dinv);
  deg_to_dinv<<<(NN + B - 1) / B, B, 0, stream>>>(dinv);

  // ---- layer 0: h0 = x @ W0 ; agg = scatter(h0) ; sigmoid(agg + b0) ----
  gemm_wmma_f32<<<NN / 16, (128 / 16) * 32, 0, stream>>>(x, W0, bufA, 128, 128);

  {
    size_t tn = (size_t)NN * 128;
    self_loop_init<<<(unsigned)((tn + B - 1) / B), B, 0, stream>>>(bufA, dinv, bufB, 128);
    size_t te = (size_t)NE * 128;
    edge_scatter<<<(unsigned)((te + B - 1) / B), B, 0, stream>>>(bufA, dinv, src, dst, bufB, 128, 7);
    bias_act<<<(unsigned)((tn + B - 1) / B), B, 0, stream>>>(bufB, b0, 128, 1);
  }

  // ---- layer 1: h1 = agg @ W1 ; out = scatter(h1) + b1 ----
  gemm_wmma_f32<<<NN / 16, (64 / 16) * 32, 0, stream>>>(bufB, W1, bufA, 128, 64);

  {
    size_t tn = (size_t)NN * 64;
    self_loop_init<<<(unsigned)((tn + B - 1) / B), B, 0, stream>>>(bufA, dinv, out, 64);
    size_t te = (size_t)NE * 64;
    edge_scatter<<<(unsigned)((te + B - 1) / B), B, 0, stream>>>(bufA, dinv, src, dst, out, 64, 6);
    bias_act<<<(unsigned)((tn + B - 1) / B), B, 0, stream>>>(out, b1, 64, 0);
  }
}